// BPaCoplusLoss_8306466750561
// MI455X (gfx1250) — compile-verified
//
#include <hip/hip_runtime.h>
#include <math.h>

// ---------------- problem constants (from reference) ----------------
#define KQ     32768
#define CCLS   100
#define BN     1024
#define DDIM   128
#define NCOL   (BN + KQ)          // 33792 feature columns
#define NTILE  (NCOL / 16)        // 2112 column tiles
#define ALPHA_ 0.05f
#define INV_T  14.28571428571428571f   // 1/0.07 ; also the analytic row-max shift
#define EPS_F  1e-12f

// ---------------- workspace layout (float offsets) ----------------
#define WS_CNT    0                       // [C]   counts over all B+K labels
#define WS_CNT1   128                     // [C]   counts over first B labels (cls_count1 = this+1)
#define WS_D1     256                     // [B]   branch1 denom partials (feature cols)
#define WS_N1     (256 + 1024)            // [B]   branch1 masked-logit partials
#define WS_D2     (256 + 2048)            // [B]   branch2 denom partials (feature cols)
#define WS_N2     (256 + 3072)            // [B]   branch2 masked-logit partials (unscaled by alpha)
#define WS_LOSS   (256 + 4096)            // [2]   loss1 sum, loss2 sum
#define WS_ZEND   (WS_LOSS + 2)
#define WS_INV0   4608                    // [NCOL] 1/cnt[lab[j]]
#define WS_INV1   (4608 + NCOL)           // [NCOL] 1/(cnt[lab[j]] - ALPHA)
#define WS_INV0B  (4608 + 2*NCOL)         // [B]    1/(cnt1[lab[j]]+1)
#define WS_INV1B  (4608 + 2*NCOL + 1024)  // [B]    1/(cnt1[lab[j]])

typedef __attribute__((ext_vector_type(2))) float v2f;
typedef __attribute__((ext_vector_type(8))) float v8f;

// ---------------- kernel 0: zero the accumulated ws region ----------------
__global__ void k_zero(float* __restrict__ ws, int n) {
  int i = blockIdx.x * blockDim.x + threadIdx.x;
  if (i < n) ws[i] = 0.0f;
}

// ---------------- kernel 1: label histograms ----------------
__global__ void k_hist(const int* __restrict__ lab, float* __restrict__ ws) {
  int j = blockIdx.x * blockDim.x + threadIdx.x;
  if (j < NCOL) {
    atomicAdd(ws + WS_CNT + lab[j], 1.0f);
    if (j < BN) atomicAdd(ws + WS_CNT1 + lab[j], 1.0f);
  }
}

// ---------------- kernel 2: per-column reciprocal weights ----------------
__global__ void k_weights(const int* __restrict__ lab, float* __restrict__ ws) {
  int j = blockIdx.x * blockDim.x + threadIdx.x;
  if (j < NCOL) {
    int l = lab[j];
    float c = ws[WS_CNT + l];                 // cls_count
    ws[WS_INV0 + j] = 1.0f / c;               // different class
    ws[WS_INV1 + j] = 1.0f / (c - ALPHA_);    // same class (mask value ALPHA)
    if (j < BN) {
      float c1 = ws[WS_CNT1 + l] + 1.0f;      // cls_count1 = batch count + 1 (center)
      ws[WS_INV0B + j] = 1.0f / c1;
      ws[WS_INV1B + j] = 1.0f / (c1 - 1.0f);  // same class (mask value 1)
    }
  }
}

// ---------------- kernel 3: fused WMMA GEMM + streaming softmax epilogue ----
// grid = (BN/16, NTILE/32), block = 128 threads = 4 waves.
// Each wave: fixed 16-row tile, 8 column tiles; D=128 contracted with 32
// v_wmma_f32_16x16x4_f32 per tile. Epilogue accumulates (per row):
//   d2 += exp((g-1)/T)/w2   over all j != i            (branch 2 feature cols)
//   n2 += (g-1)/T           over same-class j != i
//   d1,n1 same but with branch-1 weights, only for j < B
__global__ __launch_bounds__(128) void k_gemm(const float* __restrict__ feat,
                                              const int* __restrict__ lab,
                                              float* __restrict__ ws) {
  const int lane = threadIdx.x & 31;
  const int wave = threadIdx.x >> 5;
  const int lo = lane & 15;      // column index within tile (B/C/D layout)
  const int hi = lane >> 4;      // half-wave selector
  const int rowbase = blockIdx.x * 16;

  // row labels for the 8 M-rows this lane's C-fragments correspond to
  int rlab[8];
#pragma unroll
  for (int v = 0; v < 8; ++v) rlab[v] = lab[rowbase + v + 8 * hi];

  // A-panel: 16 rows x 128 K, resident in 64 VGPRs.
  // A 16x4 layout: lanes 0-15 = M; VGPR0 = K{0|2}, VGPR1 = K{1|3} -> lane holds
  // consecutive K pair starting at 4s + 2*hi  => one b64 load per K-step.
  const float* Ap = feat + (size_t)(rowbase + lo) * DDIM + 2 * hi;
  v2f a[32];
#pragma unroll
  for (int s = 0; s < 32; ++s) a[s] = *(const v2f*)(Ap + 4 * s);

  float d2a[8] = {0,0,0,0,0,0,0,0}, n2a[8] = {0,0,0,0,0,0,0,0};
  float d1a[8] = {0,0,0,0,0,0,0,0}, n1a[8] = {0,0,0,0,0,0,0,0};

  const float* inv0  = ws + WS_INV0;
  const float* inv1  = ws + WS_INV1;
  const float* inv0b = ws + WS_INV0B;
  const float* inv1b = ws + WS_INV1B;

  for (int t = 0; t < 8; ++t) {
    const int tile = blockIdx.y * 32 + t * 4 + wave;   // [0, NTILE)
    const int colbase = tile * 16;
    const int gcol = colbase + lo;
    // B-panel fragment loads mirror A (B 4x16: VGPR0 = K{0|2}, VGPR1 = K{1|3})
    const float* Bp = feat + (size_t)gcol * DDIM + 2 * hi;

    v8f c = {};
#pragma unroll
    for (int s = 0; s < 32; ++s) {
      v2f b = *(const v2f*)(Bp + 4 * s);
      c = __builtin_amdgcn_wmma_f32_16x16x4_f32(
              /*neg_a=*/false, a[s], /*neg_b=*/false, b,
              /*c_mod=*/(short)0, c, /*reuse_a=*/false, /*reuse_b=*/false);
    }

    const int   jlab = lab[gcol];
    const float i0 = inv0[gcol], i1 = inv1[gcol];
    const bool  inB = (colbase < BN);                 // whole tile on one side
    float i0b = 0.0f, i1b = 0.0f;
    if (inB) { i0b = inv0b[gcol]; i1b = inv1b[gcol]; }

#pragma unroll
    for (int v = 0; v < 8; ++v) {
      const int grow = rowbase + v + 8 * hi;
      const float sh = (c[v] - 1.0f) * INV_T;         // shift-invariant logit
      const float e  = expf(sh);
      const bool nd   = (grow != gcol);               // logits_mask (diag)
      const bool same = (rlab[v] == jlab);
      d2a[v] += nd ? (e * (same ? i1 : i0)) : 0.0f;
      const float msk = (nd && same) ? sh : 0.0f;
      n2a[v] += msk;
      if (inB) {
        d1a[v] += nd ? (e * (same ? i1b : i0b)) : 0.0f;
        n1a[v] += msk;                                 // same mask, same logits
      }
    }
  }

  // reduce each row's partial across the 16 lanes of its half, then atomics
#pragma unroll
  for (int v = 0; v < 8; ++v) {
    float x0 = d2a[v], x1 = n2a[v], x2 = d1a[v], x3 = n1a[v];
#pragma unroll
    for (int off = 8; off >= 1; off >>= 1) {   // xor 8,4,2,1 stays inside half
      x0 += __shfl_xor(x0, off, 32);
      x1 += __shfl_xor(x1, off, 32);
      x2 += __shfl_xor(x2, off, 32);
      x3 += __shfl_xor(x3, off, 32);
    }
    if (lo == 0) {
      const int r = rowbase + v + 8 * hi;
      atomicAdd(ws + WS_D2 + r, x0);
      atomicAdd(ws + WS_N2 + r, x1);
      atomicAdd(ws + WS_D1 + r, x2);
      atomicAdd(ws + WS_N1 + r, x3);
    }
  }
}

// ---------------- kernel 4: per-row finalize ----------------
// One block per anchor row: sup_logits columns (branch 2) + center columns
// (branch 1, 100x128 plain FMA), combine with GEMM partials, per-row loss.
__global__ __launch_bounds__(128) void k_row(const float* __restrict__ feat,
                                             const float* __restrict__ sup,
                                             const float* __restrict__ cen,
                                             const int* __restrict__ lab,
                                             float* __restrict__ ws) {
  __shared__ float fsh[DDIM];
  __shared__ float sred[128];
  const int i = blockIdx.x;
  const int t = threadIdx.x;
  fsh[t] = feat[(size_t)i * DDIM + t];
  __syncthreads();
  const int lbl = lab[i];

  float d1c = 0.0f, d2c = 0.0f, n1c = 0.0f, n2c = 0.0f;
  if (t < CCLS) {
    // branch 2: sup_logits column t (lmask=1, weight = cnt[t] - onehot)
    const float sl = sup[i * CCLS + t];
    const float w2 = ws[WS_CNT + t] - ((t == lbl) ? 1.0f : 0.0f);
    d2c = expf(sl - INV_T) / w2;
    if (t == lbl) n2c = sl - INV_T;
    // branch 1: center column t (j = B+t, never diagonal)
    const float* cp = cen + (size_t)t * DDIM;
    float dot = 0.0f;
    for (int k = 0; k < DDIM; ++k) dot += fsh[k] * cp[k];
    const float sh = (dot - 1.0f) * INV_T;
    const float w1 = (ws[WS_CNT1 + t] + 1.0f) - ((t == lbl) ? 1.0f : 0.0f);
    d1c = expf(sh) / w1;
    if (t == lbl) n1c = sh;
  }

  // block reductions (all threads participate uniformly)
  float vals[4] = {d1c, d2c, n1c, n2c};
  float red[4];
#pragma unroll
  for (int q = 0; q < 4; ++q) {
    sred[t] = vals[q];
    __syncthreads();
    for (int s = 64; s >= 1; s >>= 1) {
      if (t < s) sred[t] += sred[t + s];
      __syncthreads();
    }
    red[q] = sred[0];
    __syncthreads();
  }

  if (t == 0) {
    const float cnt   = ws[WS_CNT + lbl];    // cls_count[lbl]  (>=1: row i itself)
    const float cnt1b = ws[WS_CNT1 + lbl];   // cls_count1[lbl]-1 (>=1)
    // branch 1
    const float d1 = ws[WS_D1 + i] + red[0];
    const float n1 = ws[WS_N1 + i] + red[2];
    const float m1 = cnt1b;                  // mask1.sum(1)
    const float loss1 = -(n1 - m1 * logf(d1 + EPS_F)) / m1;
    // branch 2
    const float d2 = ws[WS_D2 + i] + red[1];
    const float n2 = ALPHA_ * ws[WS_N2 + i] + red[3];
    const float m2 = 1.0f + ALPHA_ * (cnt - 1.0f);   // mask_full.sum(1)
    const float loss2 = -(n2 - m2 * logf(d2 + EPS_F)) / m2;
    atomicAdd(ws + WS_LOSS + 0, loss1 * (1.0f / BN));
    atomicAdd(ws + WS_LOSS + 1, loss2 * (1.0f / BN));
  }
}

// ---------------- kernel 5: write scalar ----------------
__global__ void k_final(const float* __restrict__ ws, float* __restrict__ out) {
  out[0] = ws[WS_LOSS + 0] + ws[WS_LOSS + 1];
}

extern "C" void kernel_launch(void* const* d_in, const int* in_sizes, int n_in,
                              void* d_out, int out_size, void* d_ws, size_t ws_size,
                              hipStream_t stream) {
  const float* feat = (const float*)d_in[0];   // [B+K, D]
  const float* sup  = (const float*)d_in[1];   // [B, C]
  const float* cen  = (const float*)d_in[2];   // [C, D]
  const int*   lab  = (const int*)d_in[3];     // [B+K]
  float* out = (float*)d_out;
  float* ws  = (float*)d_ws;

  k_zero<<<(WS_ZEND + 255) / 256, 256, 0, stream>>>(ws, WS_ZEND);
  k_hist<<<(NCOL + 255) / 256, 256, 0, stream>>>(lab, ws);
  k_weights<<<(NCOL + 255) / 256, 256, 0, stream>>>(lab, ws);

  dim3 g2(BN / 16, NTILE / 32);                // 64 x 66 blocks, 4 waves each
  k_gemm<<<g2, 128, 0, stream>>>(feat, lab, ws);

  k_row<<<BN, 128, 0, stream>>>(feat, sup, cen, lab, ws);
  k_final<<<1, 1, 0, stream>>>(ws, out);
}